// FpsKnnGroup_43800076485391
// MI455X (gfx1250) — compile-verified
//
#include <hip/hip_runtime.h>

typedef __attribute__((ext_vector_type(2))) float v2f;
typedef __attribute__((ext_vector_type(8))) float v8f;

#define NB    16      // batches
#define NPT   8192    // points per batch
#define CF    64      // feature channels
#define NS    512     // fps samples
#define NK    32      // knn k

// ---------------------------------------------------------------------------
// Kernel 1: farthest point sampling. One block (512 threads) per batch.
// 16 points per thread held in registers; LDS tree argmax with
// first-occurrence (lowest-index) tie-breaking to match jnp.argmax.
// ---------------------------------------------------------------------------
__global__ __launch_bounds__(512) void fps_kernel(const float* __restrict__ xyz,
                                                  int* __restrict__ idx_out) {
  __shared__ float rv[512];
  __shared__ int   ri[512];
  __shared__ float cc[3];

  const int b = blockIdx.x;
  const int t = threadIdx.x;
  const float* xb = xyz + (size_t)b * NPT * 3;

  float px[16], py[16], pz[16], md[16];
#pragma unroll
  for (int i = 0; i < 16; ++i) {
    const int n = t + 512 * i;
    px[i] = xb[n * 3 + 0];
    py[i] = xb[n * 3 + 1];
    pz[i] = xb[n * 3 + 2];
    md[i] = 1e10f;
  }

  // initial point: argmax of squared norm
  float bv = -1.0f;
  int   bi = 0;
#pragma unroll
  for (int i = 0; i < 16; ++i) {
    const float v = px[i] * px[i] + py[i] * py[i] + pz[i] * pz[i];
    if (v > bv) { bv = v; bi = t + 512 * i; }
  }
  rv[t] = bv; ri[t] = bi;
  __syncthreads();
  for (int s = 256; s > 0; s >>= 1) {
    if (t < s) {
      const float ov = rv[t + s]; const int oi = ri[t + s];
      if (ov > rv[t] || (ov == rv[t] && oi < ri[t])) { rv[t] = ov; ri[t] = oi; }
    }
    __syncthreads();
  }
  int cur = ri[0];
  __syncthreads();

  for (int it = 0; it < NS; ++it) {
    if (t == 0) {
      idx_out[b * NS + it] = cur;
      cc[0] = xb[cur * 3 + 0];
      cc[1] = xb[cur * 3 + 1];
      cc[2] = xb[cur * 3 + 2];
    }
    __syncthreads();
    const float cx = cc[0], cy = cc[1], cz = cc[2];
    bv = -1.0f; bi = 0;
#pragma unroll
    for (int i = 0; i < 16; ++i) {
      const float dx = px[i] - cx, dy = py[i] - cy, dz = pz[i] - cz;
      const float d = dx * dx + dy * dy + dz * dz;
      float m = md[i];
      m = fminf(m, d);
      md[i] = m;
      if (m > bv) { bv = m; bi = t + 512 * i; }
    }
    __syncthreads();
    rv[t] = bv; ri[t] = bi;
    __syncthreads();
    for (int s = 256; s > 0; s >>= 1) {
      if (t < s) {
        const float ov = rv[t + s]; const int oi = ri[t + s];
        if (ov > rv[t] || (ov == rv[t] && oi < ri[t])) { rv[t] = ov; ri[t] = oi; }
      }
      __syncthreads();
    }
    cur = ri[0];
    __syncthreads();
  }
}

// ---------------------------------------------------------------------------
// Kernel 2: gather lc_xyz [B,S,3] and lc_feat [B,S,64] from fps indices.
// ---------------------------------------------------------------------------
__global__ void gather_lc_kernel(const float* __restrict__ xyz,
                                 const float* __restrict__ feat,
                                 const int* __restrict__ fps_idx,
                                 float* __restrict__ lc_xyz,
                                 float* __restrict__ lc_feat) {
  const int e = blockIdx.x * blockDim.x + threadIdx.x;
  const int total = NB * NS * (3 + CF);
  if (e >= total) return;
  const int bs = e / (3 + CF);
  const int c  = e - bs * (3 + CF);
  const int b  = bs / NS;
  const int nid = fps_idx[bs];
  if (c < 3) {
    lc_xyz[bs * 3 + c] = xyz[((size_t)b * NPT + nid) * 3 + c];
  } else {
    lc_feat[(size_t)bs * CF + (c - 3)] = feat[((size_t)b * NPT + nid) * CF + (c - 3)];
  }
}

// ---------------------------------------------------------------------------
// Kernel 3: KNN via fp32 WMMA distance tiles.
// One wave (32 threads) per tile of 16 queries.  Per 64-ref macro-chunk:
// 4 back-to-back V_WMMA_F32_16X16X4_F32 with
//   A row m = (qx,qy,qz,1)
//   B col n = (-2rx,-2ry,-2rz,||r||^2)
//   C[m][n] = ||q_m||^2
// -> D[m][n] = ||q||^2 - 2 q.r + ||r||^2  (exact reference formulation),
// then one LDS dump + one 64-candidate scan per query (lanes 0..15) into a
// sorted top-32 (ascending dist, lower index first on ties -> lax.top_k).
// ---------------------------------------------------------------------------
__global__ __launch_bounds__(32) void knn_kernel(const float* __restrict__ xyz,
                                                 const int* __restrict__ fps_idx,
                                                 int* __restrict__ nn_idx) {
  __shared__ float qs[16];
  __shared__ float dist[16 * 64];
  __shared__ float top_d[16][NK];
  __shared__ int   top_i[16][NK];

  const int blk = blockIdx.x;
  const int b   = blk >> 5;            // 32 tiles per batch
  const int s0  = (blk & 31) * 16;
  const int l   = threadIdx.x;
  const int mq  = l & 15;
  const bool hi = (l >= 16);
  const int mbase = hi ? 8 : 0;
  const float* xb = xyz + (size_t)b * NPT * 3;

  const int qid = fps_idx[b * NS + s0 + mq];
  const float qx = xb[qid * 3 + 0];
  const float qy = xb[qid * 3 + 1];
  const float qz = xb[qid * 3 + 2];
  if (!hi) {
    qs[mq] = qx * qx + qy * qy + qz * qz;
    for (int k = 0; k < NK; ++k) { top_d[mq][k] = 3.0e38f; top_i[mq][k] = 0; }
  }
  __syncthreads();

  // A matrix (16x4 f32): lanes 0-15 hold (K=0,K=1)=(qx,qy); lanes 16-31 (K=2,K=3)=(qz,1)
  v2f a;
  a.x = hi ? qz : qx;
  a.y = hi ? 1.0f : qy;

  // C matrix: VGPR j, lanes 0-15 -> M=j ; lanes 16-31 -> M=j+8 ; broadcast ||q_M||^2
  v8f cacc;
#pragma unroll
  for (int j = 0; j < 8; ++j) cacc[j] = qs[mbase + j];

  float thr = 3.0e38f;

  for (int ch = 0; ch < NPT / 64; ++ch) {
    const int base = ch * 64;
    v8f d[4];
#pragma unroll
    for (int sub = 0; sub < 4; ++sub) {
      const int rid = base + sub * 16 + mq;
      const float rx = xb[rid * 3 + 0];
      const float ry = xb[rid * 3 + 1];
      const float rz = xb[rid * 3 + 2];
      // B matrix (4x16 f32): v0 = (K=0 lanes0-15, K=2 lanes16-31); v1 = (K=1, K=3)
      v2f bm;
      bm.x = hi ? (-2.0f * rz) : (-2.0f * rx);
      bm.y = hi ? (rx * rx + ry * ry + rz * rz) : (-2.0f * ry);
      d[sub] = __builtin_amdgcn_wmma_f32_16x16x4_f32(
          /*neg_a=*/false, a, /*neg_b=*/false, bm,
          /*c_mod=*/(short)0, cacc, /*reuse_a=*/false, /*reuse_b=*/false);
    }

    __syncthreads();  // prior selection reads of dist[] complete (WAR)
#pragma unroll
    for (int sub = 0; sub < 4; ++sub)
#pragma unroll
      for (int j = 0; j < 8; ++j)
        dist[(mbase + j) * 64 + sub * 16 + mq] = d[sub][j];
    __syncthreads();

    if (!hi) {
#pragma unroll 1
      for (int r = 0; r < 64; ++r) {
        const float dv = fmaxf(dist[mq * 64 + r], 0.0f);  // reference clips at 0
        if (dv < thr) {
          int j = NK - 1;
          while (j > 0 && top_d[mq][j - 1] > dv) {
            top_d[mq][j] = top_d[mq][j - 1];
            top_i[mq][j] = top_i[mq][j - 1];
            --j;
          }
          top_d[mq][j] = dv;
          top_i[mq][j] = base + r;
          thr = top_d[mq][NK - 1];
        }
      }
    }
  }
  __syncthreads();
  if (!hi) {
    for (int k = 0; k < NK; ++k)
      nn_idx[((size_t)(b * NS + s0 + mq)) * NK + k] = top_i[mq][k];
  }
}

// ---------------------------------------------------------------------------
// Kernel 4: gather knn_xyz [B,S,K,3] and knn_feat [B,S,K,64+3].
// One block per (b,s); neighbor ids staged in LDS, feat rows prefetched
// (global_prefetch_b8) while the block is at the barrier.
// ---------------------------------------------------------------------------
__global__ __launch_bounds__(256) void gather_knn_kernel(const float* __restrict__ xyz,
                                                         const float* __restrict__ feat,
                                                         const int* __restrict__ nn_idx,
                                                         float* __restrict__ knn_xyz,
                                                         float* __restrict__ knn_feat) {
  __shared__ int nid_s[NK];
  const int bs = blockIdx.x;
  const int b  = bs / NS;
  const int t  = threadIdx.x;
  if (t < NK) {
    const int nid = nn_idx[(size_t)bs * NK + t];
    nid_s[t] = nid;
    __builtin_prefetch(&feat[((size_t)b * NPT + nid) * CF], 0, 1);
  }
  __syncthreads();

  // knn_xyz: 32*3 = 96 elements
  if (t < NK * 3) {
    const int k = t / 3, c = t - k * 3;
    knn_xyz[(size_t)bs * (NK * 3) + t] = xyz[((size_t)b * NPT + nid_s[k]) * 3 + c];
  }
  // knn_feat: 32*67 = 2144 elements, feat channels then xyz
  const int fw = CF + 3;
  for (int e = t; e < NK * fw; e += 256) {
    const int k = e / fw, c = e - k * fw;
    const int nid = nid_s[k];
    float v;
    if (c < CF) v = feat[((size_t)b * NPT + nid) * CF + c];
    else        v = xyz[((size_t)b * NPT + nid) * 3 + (c - CF)];
    knn_feat[(size_t)bs * (NK * fw) + e] = v;
  }
}

// ---------------------------------------------------------------------------
extern "C" void kernel_launch(void* const* d_in, const int* in_sizes, int n_in,
                              void* d_out, int out_size, void* d_ws, size_t ws_size,
                              hipStream_t stream) {
  const float* xyz  = (const float*)d_in[0];   // [16, 8192, 3]
  const float* feat = (const float*)d_in[1];   // [16, 8192, 64]

  float* out      = (float*)d_out;
  float* lc_xyz   = out;                              // 16*512*3
  float* lc_feat  = lc_xyz  + (size_t)NB * NS * 3;    // 16*512*64
  float* knn_xyz  = lc_feat + (size_t)NB * NS * CF;   // 16*512*32*3
  float* knn_feat = knn_xyz + (size_t)NB * NS * NK * 3;

  int* fps_idx = (int*)d_ws;                   // 16*512
  int* nn_idx  = fps_idx + NB * NS;            // 16*512*32

  fps_kernel<<<NB, 512, 0, stream>>>(xyz, fps_idx);

  {
    const int total = NB * NS * (3 + CF);
    gather_lc_kernel<<<(total + 255) / 256, 256, 0, stream>>>(xyz, feat, fps_idx,
                                                              lc_xyz, lc_feat);
  }

  knn_kernel<<<NB * (NS / 16), 32, 0, stream>>>(xyz, fps_idx, nn_idx);

  gather_knn_kernel<<<NB * NS, 256, 0, stream>>>(xyz, feat, nn_idx, knn_xyz, knn_feat);
}